// FaceRayAttentionAdapter_81071802679604
// MI455X (gfx1250) — compile-verified
//
#include <hip/hip_runtime.h>
#include <hip/hip_bf16.h>

// ---------------- problem constants ----------------
#define EMBED 1024
#define ADIM  1024
#define NH    16
#define HD    64          // ADIM / NH
#define RAY   48
#define NB    768
#define MN    8
#define FG    6
#define LL    1024
#define BB_   12
#define BSZ   2           // BB_ / FG
#define STAGES 4
#define ATT_SCALE 0.125f  // 1/sqrt(64)

typedef __attribute__((ext_vector_type(2))) float v2f;
typedef __attribute__((ext_vector_type(8))) float v8f;

__device__ __forceinline__ int iclampi(int v, int lo, int hi) {
    return v < lo ? lo : (v > hi ? hi : v);
}

// ---------------- kernel 0: index prep ----------------
__global__ __launch_bounds__(256)
void prep_indices(const int* __restrict__ bface, const int* __restrict__ btok,
                  const int* __restrict__ nface, const int* __restrict__ ntok,
                  long long* __restrict__ qrow,  long long* __restrict__ kvrow,
                  long long* __restrict__ orow,  float* __restrict__ rvalid) {
    int i = blockIdx.x * blockDim.x + threadIdx.x;
    if (i < NB) {
        int f = bface[i], tk = btok[i];
        int fc = iclampi(f, 0, FG - 1);
        int tc = iclampi(tk, 0, LL - 1);
        float v = (f == fc && tk == tc) ? 1.0f : 0.0f;
        for (int bb = 0; bb < BSZ; ++bb) {
            long long ofs = (((long long)bb * FG + fc) * LL + tc) * (long long)EMBED;
            qrow[bb * NB + i]   = ofs;
            orow[bb * NB + i]   = ofs;
            rvalid[bb * NB + i] = v;
        }
    }
    if (i < NB * MN) {
        int f = nface[i], tk = ntok[i];
        int fc = iclampi(f, 0, FG - 1);
        int tc = iclampi(tk, 0, LL - 1);
        for (int bb = 0; bb < BSZ; ++bb) {
            kvrow[bb * NB * MN + i] =
                (((long long)bb * FG + fc) * LL + tc) * (long long)EMBED;
        }
    }
}

// ---------------- kernel 1: zero output ----------------
__global__ __launch_bounds__(256)
void zero_out(float4* __restrict__ p, long long n4) {
    long long i = (long long)blockIdx.x * blockDim.x + threadIdx.x;
    if (i < n4) p[i] = make_float4(0.f, 0.f, 0.f, 0.f);
}

// ---------------- kernel 2: generic fp32 WMMA GEMM ----------------
// C[row, n0..] = (gatherA(row, :) @ W[:, n0..] + bias) * rscale[row], row scattered via crow.
// Block tile 128x64, BK=32, 8 waves in a 4(M)x2(N) grid; each wave computes 32x32
// (four 16x16 WMMA accumulators). Double-buffered LDS: global loads for tile i+1
// are issued before the WMMA stack for tile i, hiding load latency behind compute.
#define BM 128
#define BN 64
#define BK 32
#define LDA_S 36   // padded stride, float4-aligned (36*4 B % 16 == 0), staggers banks
#define LDB_S 68   // 68*4 = 272 B per row, 16B aligned

__global__ __launch_bounds__(256)
void wmma_gemm_f32(const float* __restrict__ A,
                   const long long* __restrict__ arow, int lda,
                   const float* __restrict__ W, int ldw,
                   float* __restrict__ C,
                   const long long* __restrict__ crow, int ldc,
                   const float* __restrict__ bias,
                   const float* __restrict__ rscale,
                   int K) {
    __shared__ float lA[2][BM * LDA_S];
    __shared__ float lB[2][BK * LDB_S];

    const int m0 = blockIdx.y * BM;
    const int n0 = blockIdx.x * BN;
    const int t  = threadIdx.x;

    // A-tile loader: 128 rows x 32 floats = 1024 float4; thread covers rows ar+{0,32,64,96}
    const int ar = t >> 3;
    const int ac = (t & 7) * 4;
    long long aofs[4];
    #pragma unroll
    for (int i = 0; i < 4; ++i) {
        const int row = m0 + ar + 32 * i;
        aofs[i] = arow ? arow[row] : (long long)row * lda;
    }
    // B-tile loader: 32 rows x 64 floats = 512 float4; rows br and br+16
    const int br = t >> 4;
    const int bc = (t & 15) * 4;

    // wave mapping: 8 waves -> 4 (M) x 2 (N); each wave owns a 32x32 output slab
    const int w    = t >> 5;
    const int lane = t & 31;
    const int half = lane >> 4;   // ISA: lanes 16-31 carry K+2 for A/B frags, M+8 for C
    const int ml   = lane & 15;
    const int wm   = (w >> 1) * 32;
    const int wn   = (w & 1) * 32;

    v8f c00 = {0.f,0.f,0.f,0.f,0.f,0.f,0.f,0.f};
    v8f c01 = {0.f,0.f,0.f,0.f,0.f,0.f,0.f,0.f};
    v8f c10 = {0.f,0.f,0.f,0.f,0.f,0.f,0.f,0.f};
    v8f c11 = {0.f,0.f,0.f,0.f,0.f,0.f,0.f,0.f};

    float4 areg[4], breg[2];
    auto LOAD = [&](int k0) {
        #pragma unroll
        for (int i = 0; i < 4; ++i)
            areg[i] = *(const float4*)(A + aofs[i] + k0 + ac);
        breg[0] = *(const float4*)(W + (long long)(k0 + br) * ldw + n0 + bc);
        breg[1] = *(const float4*)(W + (long long)(k0 + br + 16) * ldw + n0 + bc);
    };
    auto STORE = [&](int s) {
        #pragma unroll
        for (int i = 0; i < 4; ++i)
            *(float4*)(&lA[s][(ar + 32 * i) * LDA_S + ac]) = areg[i];
        *(float4*)(&lB[s][br * LDB_S + bc])        = breg[0];
        *(float4*)(&lB[s][(br + 16) * LDB_S + bc]) = breg[1];
    };

    LOAD(0);
    STORE(0);
    __syncthreads();

    int cur = 0;
    for (int k0 = 0; k0 < K; k0 += BK) {
        const bool more = (k0 + BK) < K;
        if (more) LOAD(k0 + BK);   // overlap with the WMMA stack below

        const float* LA = lA[cur];
        const float* LB = lB[cur];
        #pragma unroll
        for (int kk = 0; kk < BK; kk += 4) {
            const int kb = kk + half * 2;
            // A frags 16x4 (two M-tiles)
            const float* ap0 = LA + (wm + ml) * LDA_S + kb;
            const float* ap1 = LA + (wm + 16 + ml) * LDA_S + kb;
            v2f a0; a0.x = ap0[0]; a0.y = ap0[1];
            v2f a1; a1.x = ap1[0]; a1.y = ap1[1];
            // B frags 4x16 (two N-tiles)
            v2f b0, b1;
            b0.x = LB[kb * LDB_S + wn + ml];
            b0.y = LB[(kb + 1) * LDB_S + wn + ml];
            b1.x = LB[kb * LDB_S + wn + 16 + ml];
            b1.y = LB[(kb + 1) * LDB_S + wn + 16 + ml];
            c00 = __builtin_amdgcn_wmma_f32_16x16x4_f32(false, a0, false, b0, (short)0, c00, false, false);
            c01 = __builtin_amdgcn_wmma_f32_16x16x4_f32(false, a0, false, b1, (short)0, c01, false, false);
            c10 = __builtin_amdgcn_wmma_f32_16x16x4_f32(false, a1, false, b0, (short)0, c10, false, false);
            c11 = __builtin_amdgcn_wmma_f32_16x16x4_f32(false, a1, false, b1, (short)0, c11, false, false);
        }
        __syncthreads();                       // all reads of lds[cur] done
        if (more) {
            STORE(cur ^ 1);                    // fill the other buffer
            __syncthreads();                   // writes visible before next compute
            cur ^= 1;
        }
    }

    // C layout: VGPR r -> M = tile_m + half*8 + r, N = ml
    #pragma unroll
    for (int r2 = 0; r2 < 8; ++r2) {
        const int row0 = m0 + wm + half * 8 + r2;
        const int row1 = row0 + 16;
        const int col0 = n0 + wn + ml;
        const int col1 = col0 + 16;
        const float sc0 = rscale ? rscale[row0] : 1.0f;
        const float sc1 = rscale ? rscale[row1] : 1.0f;
        const long long co0 = crow ? crow[row0] : (long long)row0 * ldc;
        const long long co1 = crow ? crow[row1] : (long long)row1 * ldc;
        const float bv0 = bias ? bias[col0] : 0.0f;
        const float bv1 = bias ? bias[col1] : 0.0f;
        C[co0 + col0] = (c00[r2] + bv0) * sc0;
        C[co0 + col1] = (c01[r2] + bv1) * sc0;
        C[co1 + col0] = (c10[r2] + bv0) * sc1;
        C[co1 + col1] = (c11[r2] + bv1) * sc1;
    }
}

// ---------------- kernel 3: ray-rotated neighbor attention ----------------
// one block (256 thr) per (bb, n). qbuf: (BSZ*NB, 1024). kvbuf: (BSZ*NB*MN, 2048) [k|v].
__global__ __launch_bounds__(256)
void attn_kernel(const float* __restrict__ qbuf, const float* __restrict__ kvbuf,
                 const float* __restrict__ d_ray,
                 const int* __restrict__ bface, const int* __restrict__ bloc,
                 const int* __restrict__ bstage,
                 const int* __restrict__ nface, const int* __restrict__ ntok,
                 const int* __restrict__ nloc,  const int* __restrict__ nstage,
                 const unsigned char* __restrict__ nmask,
                 float* __restrict__ outattn) {
    const int blk = blockIdx.x;        // 0 .. BSZ*NB-1
    const int n   = blk % NB;
    const int t   = threadIdx.x;
    const int h    = t >> 4;           // head 0..15
    const int lane = t & 15;
    const int d0   = lane * 4;         // 4 consecutive head-dims per thread
    const int i3   = d0 >> 4;          // ray block index (valid when d0 < RAY)
    const int r0   = d0 & 15;

    __shared__ float sq[ADIM];         // raw q row
    __shared__ float sk[ADIM];         // raw k row (per m)
    __shared__ float svraw[ADIM];      // raw v row (per m)
    __shared__ float sv[MN * ADIM];    // rotated v, all m
    __shared__ float so[ADIM];         // attention output before inverse rotation
    __shared__ float sdq[9];
    __shared__ float sdk[MN * 9];
    __shared__ float slog[MN * NH];
    __shared__ float swt[MN * NH];
    __shared__ float smask[MN];

    if (t < 9) {
        int st = iclampi(bstage[n], 0, STAGES - 1);
        int fa = iclampi(bface[n], 0, FG - 1);
        int lo = iclampi(bloc[n], 0, LL - 1);
        sdq[t] = d_ray[(((long long)st * FG + fa) * LL + lo) * 9 + t];
    }
    if (t < MN * 9) {
        int m = t / 9, e = t % 9;
        int st = iclampi(nstage[n * MN + m], 0, STAGES - 1);
        int fa = iclampi(nface[n * MN + m], 0, FG - 1);   // ref uses max(.,0); in-range equal
        int lo = iclampi(nloc[n * MN + m], 0, LL - 1);
        sdk[t] = d_ray[(((long long)st * FG + fa) * LL + lo) * 9 + e];
    }
    if (t < MN) {
        int fa = nface[n * MN + t], tk = ntok[n * MN + t];
        bool ok = nmask[n * MN + t] && fa >= 0 && fa < FG && tk >= 0 && tk < LL;
        smask[t] = ok ? 1.0f : 0.0f;
    }
    ((float4*)sq)[t] = ((const float4*)(qbuf + (long long)blk * ADIM))[t];
    __syncthreads();

    // rotate q (first 48 of each head-dim get the 3x3; rest pass through)
    float qr[4];
    if (d0 < RAY) {
        const float a0 = sdq[i3 * 3 + 0], a1 = sdq[i3 * 3 + 1], a2 = sdq[i3 * 3 + 2];
        #pragma unroll
        for (int e = 0; e < 4; ++e) {
            const int rr = r0 + e;
            qr[e] = a0 * sq[h * HD + rr] + a1 * sq[h * HD + 16 + rr] + a2 * sq[h * HD + 32 + rr];
        }
    } else {
        #pragma unroll
        for (int e = 0; e < 4; ++e) qr[e] = sq[h * HD + d0 + e];
    }

    for (int m = 0; m < MN; ++m) {
        __syncthreads();   // previous iteration done with sk/svraw
        const float* kv = kvbuf + ((long long)blk * MN + m) * 2048;
        ((float4*)sk)[t]    = ((const float4*)kv)[t];
        ((float4*)svraw)[t] = ((const float4*)(kv + ADIM))[t];
        __syncthreads();

        float kr[4], vr[4];
        if (d0 < RAY) {
            const float b0 = sdk[m * 9 + i3 * 3 + 0];
            const float b1 = sdk[m * 9 + i3 * 3 + 1];
            const float b2 = sdk[m * 9 + i3 * 3 + 2];
            #pragma unroll
            for (int e = 0; e < 4; ++e) {
                const int rr = r0 + e;
                kr[e] = b0 * sk[h * HD + rr] + b1 * sk[h * HD + 16 + rr] + b2 * sk[h * HD + 32 + rr];
                vr[e] = b0 * svraw[h * HD + rr] + b1 * svraw[h * HD + 16 + rr] + b2 * svraw[h * HD + 32 + rr];
            }
        } else {
            #pragma unroll
            for (int e = 0; e < 4; ++e) {
                kr[e] = sk[h * HD + d0 + e];
                vr[e] = svraw[h * HD + d0 + e];
            }
        }
        #pragma unroll
        for (int e = 0; e < 4; ++e) sv[m * ADIM + h * HD + d0 + e] = vr[e];

        float part = qr[0] * kr[0] + qr[1] * kr[1] + qr[2] * kr[2] + qr[3] * kr[3];
        part += __shfl_xor(part, 1, 16);
        part += __shfl_xor(part, 2, 16);
        part += __shfl_xor(part, 4, 16);
        part += __shfl_xor(part, 8, 16);
        if (lane == 0) slog[m * NH + h] = part * ATT_SCALE;
    }
    __syncthreads();

    // masked softmax over m, per head (threads 0..15)
    if (t < NH) {
        float mx = -3.4e38f;
        #pragma unroll
        for (int m = 0; m < MN; ++m)
            if (smask[m] > 0.f) mx = fmaxf(mx, slog[m * NH + t]);
        float s = 0.f;
        #pragma unroll
        for (int m = 0; m < MN; ++m) {
            float wv = (smask[m] > 0.f) ? __expf(slog[m * NH + t] - mx) : 0.f;
            swt[m * NH + t] = wv;
            s += wv;
        }
        const float inv = (s > 0.f) ? (1.0f / s) : 0.f;
        #pragma unroll
        for (int m = 0; m < MN; ++m) swt[m * NH + t] *= inv;
    }
    __syncthreads();

    // out = sum_m w[m] * v_rot[m]
    float o[4] = {0.f, 0.f, 0.f, 0.f};
    for (int m = 0; m < MN; ++m) {
        const float wv = swt[m * NH + h];
        #pragma unroll
        for (int e = 0; e < 4; ++e) o[e] += wv * sv[m * ADIM + h * HD + d0 + e];
    }
    #pragma unroll
    for (int e = 0; e < 4; ++e) so[h * HD + d0 + e] = o[e];
    __syncthreads();

    // inverse (transposed) d_q rotation
    float oo[4];
    if (d0 < RAY) {
        const float a0 = sdq[0 * 3 + i3], a1 = sdq[1 * 3 + i3], a2 = sdq[2 * 3 + i3];
        #pragma unroll
        for (int e = 0; e < 4; ++e) {
            const int rr = r0 + e;
            oo[e] = a0 * so[h * HD + rr] + a1 * so[h * HD + 16 + rr] + a2 * so[h * HD + 32 + rr];
        }
    } else {
        #pragma unroll
        for (int e = 0; e < 4; ++e) oo[e] = o[e];
    }
    float* op = outattn + (long long)blk * ADIM + h * HD + d0;
    #pragma unroll
    for (int e = 0; e < 4; ++e) op[e] = oo[e];
}

// ---------------- host launch ----------------
extern "C" void kernel_launch(void* const* d_in, const int* in_sizes, int n_in,
                              void* d_out, int out_size, void* d_ws, size_t ws_size,
                              hipStream_t stream) {
    const float* x      = (const float*)d_in[0];
    const float* qkv_w  = (const float*)d_in[1];
    const float* out_w  = (const float*)d_in[2];
    const float* out_b  = (const float*)d_in[3];
    const float* d_ray  = (const float*)d_in[4];
    const int*   bface  = (const int*)d_in[5];
    const int*   btok   = (const int*)d_in[6];
    const int*   bloc   = (const int*)d_in[7];
    const int*   bstage = (const int*)d_in[8];
    const int*   nface  = (const int*)d_in[9];
    const int*   ntok   = (const int*)d_in[10];
    const int*   nloc   = (const int*)d_in[11];
    const int*   nstage = (const int*)d_in[12];
    const unsigned char* nmask = (const unsigned char*)d_in[13];
    float* out = (float*)d_out;

    // workspace layout (all 256B aligned)
    char* ws = (char*)d_ws;
    size_t off = 0;
    float*     qbuf   = (float*)(ws + off); off += (size_t)BSZ * NB * ADIM * 4;            // 6 MB
    float*     kvbuf  = (float*)(ws + off); off += (size_t)BSZ * NB * MN * 2048 * 4;       // 100 MB
    float*     oattn  = (float*)(ws + off); off += (size_t)BSZ * NB * ADIM * 4;            // 6 MB
    long long* qrow   = (long long*)(ws + off); off += (size_t)BSZ * NB * 8;
    long long* kvrow  = (long long*)(ws + off); off += (size_t)BSZ * NB * MN * 8;
    long long* orow   = (long long*)(ws + off); off += (size_t)BSZ * NB * 8;
    float*     rvalid = (float*)(ws + off); off += (size_t)BSZ * NB * 4;

    // 0) indices
    prep_indices<<<(NB * MN + 255) / 256, 256, 0, stream>>>(bface, btok, nface, ntok,
                                                            qrow, kvrow, orow, rvalid);
    // 1) zero the scatter target
    {
        long long n4 = (long long)BB_ * LL * EMBED / 4;
        zero_out<<<(unsigned)((n4 + 255) / 256), 256, 0, stream>>>((float4*)out, n4);
    }
    // 2) q = gather(x) @ qkv_w[:, :1024]   (M=1536, N=1024, K=1024)
    wmma_gemm_f32<<<dim3(ADIM / BN, (BSZ * NB) / BM), 256, 0, stream>>>(
        x, qrow, 0, qkv_w, 3 * ADIM, qbuf, nullptr, ADIM, nullptr, nullptr, EMBED);
    // 3) kv = gather(x) @ qkv_w[:, 1024:3072]  (M=12288, N=2048, K=1024)
    wmma_gemm_f32<<<dim3(2048 / BN, (BSZ * NB * MN) / BM), 256, 0, stream>>>(
        x, kvrow, 0, qkv_w + ADIM, 3 * ADIM, kvbuf, nullptr, 2048, nullptr, nullptr, EMBED);
    // 4) ray-rotated neighbor attention
    attn_kernel<<<BSZ * NB, 256, 0, stream>>>(qbuf, kvbuf, d_ray,
                                              bface, bloc, bstage,
                                              nface, ntok, nloc, nstage, nmask, oattn);
    // 5) out = (oattn @ out_w + out_b) * valid, scattered into zeroed output
    wmma_gemm_f32<<<dim3(EMBED / BN, (BSZ * NB) / BM), 256, 0, stream>>>(
        oattn, nullptr, ADIM, out_w, EMBED, out, orow, 0, out_b, rvalid, ADIM);
}